// VectorQuantizer_32736240730480
// MI455X (gfx1250) — compile-verified
//
#include <hip/hip_runtime.h>
#include <float.h>

// Problem constants
constexpr int NUM_EMB = 8192;   // K
constexpr int EDIM    = 256;    // D
constexpr int NB      = 16;     // batch
constexpr int HH      = 32;
constexpr int WW      = 32;
constexpr int NVEC    = NB * HH * WW;          // 16384 rows
constexpr int Q_ELEMS = NB * EDIM * HH * WW;   // 4194304
constexpr int MTILE   = 64;                    // rows per workgroup (4 WMMA tiles/wave)
constexpr int NBLK    = NVEC / MTILE;          // 256 workgroups
constexpr int ASTRIDE = EDIM + 4;              // 260: bank-conflict-free b64 A loads

typedef float v2f __attribute__((ext_vector_type(2)));
typedef float v8f __attribute__((ext_vector_type(8)));

// ---------------------------------------------------------------- ee = ||e||^2
__global__ __launch_bounds__(256) void vq_ee_kernel(const float* __restrict__ emb,
                                                    float* __restrict__ ee) {
    const int wave = threadIdx.x >> 5;
    const int lane = threadIdx.x & 31;
    const int row  = blockIdx.x * 8 + wave;          // 1024 blocks * 8 waves
    const float* r = emb + (size_t)row * EDIM;
    float s = 0.0f;
    #pragma unroll
    for (int i = 0; i < EDIM / 32; ++i) {
        float v = r[lane + 32 * i];
        s = fmaf(v, v, s);
    }
    #pragma unroll
    for (int off = 16; off >= 1; off >>= 1) s += __shfl_xor(s, off, 32);
    if (lane == 0) ee[row] = s;
}

// ---------------------------------------------------------------- main kernel
__global__ __launch_bounds__(128) void vq_main_kernel(const float* __restrict__ in,
                                                      const float* __restrict__ emb,
                                                      const float* __restrict__ ee,
                                                      float* __restrict__ out,
                                                      float* __restrict__ partials) {
    __shared__ __align__(16) float As[MTILE * ASTRIDE];   // 64x256 fp32 A tile (padded)
    __shared__ float xx[MTILE];                           // ||x||^2 per row
    __shared__ float red_d[MTILE * 4];                    // per-(row, wave) best dist
    __shared__ int   red_i[MTILE * 4];                    // matching argmin index
    __shared__ int   sel[MTILE];                          // final index per row

    const int t  = threadIdx.x;
    const int n0 = blockIdx.x * MTILE;                    // first flat row of this tile
    const int b  = n0 >> 10;                              // n0 / (H*W)
    const int h0 = (n0 & 1023) >> 5;                      // rows m: h=h0+(m>>5), w=m&31
    const size_t ibase = (size_t)b * (EDIM * HH * WW) + (size_t)h0 * WW;
    // in[b][c][h][w] for row m, dim c == in[ibase + c*1024 + m]   (m = 0..63)

    // ---- stage A tile (coalesced 64-wide over m)
    {
        const int m = t & 63;
        for (int c = t >> 6; c < EDIM; c += 2)
            As[m * ASTRIDE + c] = in[ibase + (size_t)c * (HH * WW) + m];
    }
    __syncthreads();

    // ---- per-row ||x||^2
    if (t < MTILE) {
        const float* r = &As[t * ASTRIDE];
        float s = 0.0f;
        for (int c = 0; c < EDIM; ++c) s = fmaf(r[c], r[c], s);
        xx[t] = s;
    }
    __syncthreads();

    const int wave = t >> 5;
    const int lane = t & 31;
    const int j    = lane & 15;       // column within this wave's 16-codeword strip
    const int hsel = lane >> 4;       // fp32 WMMA layout: K-pair {0,1} vs {2,3}

    // cached ||x||^2 for the 8 C-rows each lane owns in each of the 4 M-subtiles
    float xxc[4][8];
    #pragma unroll
    for (int T = 0; T < 4; ++T)
        #pragma unroll
        for (int v = 0; v < 8; ++v)
            xxc[T][v] = xx[T * 16 + v + 8 * hsel];

    float bd[4][8];
    int   bi[4][8];
    #pragma unroll
    for (int T = 0; T < 4; ++T)
        #pragma unroll
        for (int v = 0; v < 8; ++v) { bd[T][v] = FLT_MAX; bi[T][v] = 0x7fffffff; }

    const float* AsJ[4];
    #pragma unroll
    for (int T = 0; T < 4; ++T)
        AsJ[T] = &As[(T * 16 + j) * ASTRIDE + 2 * hsel];

    for (int kb = 0; kb < NUM_EMB; kb += 64) {
        const int kcol = kb + wave * 16 + j;              // this lane's codeword
        const float* brow = emb + (size_t)kcol * EDIM + 2 * hsel;

        v8f acc0 = {}, acc1 = {}, acc2 = {}, acc3 = {};
        #pragma unroll 16
        for (int dd = 0; dd < EDIM / 4; ++dd) {
            const int d0 = dd * 4;
            v2f bv = *(const v2f*)(brow + d0);            // one B load feeds 4 WMMAs
            v2f a0 = *(const v2f*)(AsJ[0] + d0);
            v2f a1 = *(const v2f*)(AsJ[1] + d0);
            v2f a2 = *(const v2f*)(AsJ[2] + d0);
            v2f a3 = *(const v2f*)(AsJ[3] + d0);
            acc0 = __builtin_amdgcn_wmma_f32_16x16x4_f32(false, a0, false, bv, (short)0, acc0, false, false);
            acc1 = __builtin_amdgcn_wmma_f32_16x16x4_f32(false, a1, false, bv, (short)0, acc1, false, false);
            acc2 = __builtin_amdgcn_wmma_f32_16x16x4_f32(false, a2, false, bv, (short)0, acc2, false, false);
            acc3 = __builtin_amdgcn_wmma_f32_16x16x4_f32(false, a3, false, bv, (short)0, acc3, false, false);
        }

        // Per-lane kcol strictly increases across chunks, so strict '<' alone
        // preserves first-occurrence argmin semantics here (lean epilogue).
        const float eek = ee[kcol];
        #pragma unroll
        for (int v = 0; v < 8; ++v) {
            float d0v = fmaf(-2.0f, acc0[v], xxc[0][v] + eek);
            float d1v = fmaf(-2.0f, acc1[v], xxc[1][v] + eek);
            float d2v = fmaf(-2.0f, acc2[v], xxc[2][v] + eek);
            float d3v = fmaf(-2.0f, acc3[v], xxc[3][v] + eek);
            if (d0v < bd[0][v]) { bd[0][v] = d0v; bi[0][v] = kcol; }
            if (d1v < bd[1][v]) { bd[1][v] = d1v; bi[1][v] = kcol; }
            if (d2v < bd[2][v]) { bd[2][v] = d2v; bi[2][v] = kcol; }
            if (d3v < bd[3][v]) { bd[3][v] = d3v; bi[3][v] = kcol; }
        }
    }

    // ---- in-register argmin reduction over the 16 columns (order-independent
    //      min + lowest-index tie break => deterministic), then spill per wave
    #pragma unroll
    for (int T = 0; T < 4; ++T) {
        #pragma unroll
        for (int v = 0; v < 8; ++v) {
            float d = bd[T][v];
            int   i = bi[T][v];
            #pragma unroll
            for (int mask = 1; mask < 16; mask <<= 1) {
                float od = __shfl_xor(d, mask, 32);
                int   oi = __shfl_xor(i, mask, 32);
                if (od < d || (od == d && oi < i)) { d = od; i = oi; }
            }
            if (j == 0) {
                const int r = T * 16 + v + 8 * hsel;
                red_d[r * 4 + wave] = d;
                red_i[r * 4 + wave] = i;
            }
        }
    }
    __syncthreads();

    // ---- final argmin across the 4 waves, emit index output
    if (t < MTILE) {
        float best = FLT_MAX;
        int   bidx = 0x7fffffff;
        #pragma unroll
        for (int s = 0; s < 4; ++s) {
            float d = red_d[t * 4 + s];
            int   i = red_i[t * 4 + s];
            if (d < best || (d == best && i < bidx)) { best = d; bidx = i; }
        }
        sel[t] = bidx;
        out[Q_ELEMS + 1 + n0 + t] = (float)bidx;   // idx output (as float)
    }
    __syncthreads();

    // ---- write q_st = x + (q - x), accumulate loss partial
    float lsum = 0.0f;
    {
        const int m = t & 63;
        const float* qrow = emb + (size_t)sel[m] * EDIM;
        for (int c = t >> 6; c < EDIM; c += 2) {
            float x    = As[m * ASTRIDE + c];
            float diff = qrow[c] - x;
            out[ibase + (size_t)c * (HH * WW) + m] = x + diff;
            lsum = fmaf(diff, diff, lsum);
        }
    }
    red_d[t] = lsum;           // red_d is free again (post-barrier), 256 slots >= 128
    __syncthreads();
    if (t == 0) {
        float s = 0.0f;
        for (int i = 0; i < 128; ++i) s += red_d[i];
        partials[blockIdx.x] = s;   // deterministic: no float atomics
    }
}

// ---------------------------------------------------------------- loss finalize
__global__ void vq_finalize_kernel(const float* __restrict__ partials,
                                   float* __restrict__ out) {
    float s = 0.0f;
    for (int i = 0; i < NBLK; ++i) s += partials[i];
    float m = s * (1.0f / (float)Q_ELEMS);
    out[Q_ELEMS] = m + 0.25f * m;   // q_latent + commitment * e_latent
}

extern "C" void kernel_launch(void* const* d_in, const int* in_sizes, int n_in,
                              void* d_out, int out_size, void* d_ws, size_t ws_size,
                              hipStream_t stream) {
    const float* in  = (const float*)d_in[0];   // [16,256,32,32] fp32
    const float* emb = (const float*)d_in[1];   // [8192,256] fp32
    float* out = (float*)d_out;                 // [q_st | loss | idx]
    float* ws  = (float*)d_ws;
    float* ee       = ws;                       // 8192 floats
    float* partials = ws + NUM_EMB;             // 256 floats

    vq_ee_kernel<<<NUM_EMB / 8, 256, 0, stream>>>(emb, ee);
    vq_main_kernel<<<NBLK, 128, 0, stream>>>(in, emb, ee, out, partials);
    vq_finalize_kernel<<<1, 1, 0, stream>>>(partials, out);
}